// NonMaximumSuppression_11312943858261
// MI455X (gfx1250) — compile-verified
//
#include <hip/hip_runtime.h>
#include <stdint.h>

// ---------------- problem constants (from reference) ----------------
#define NUM_CLASSES 81
#define MAX_BOXES_K 300
#define NEGV        (-1000000000.0f)

// ---------------- tuning ----------------
constexpr int NMS_BLOCKS = 120;   // persistent workgroups; must all be co-resident
constexpr int NMS_TPB    = 256;   // 8 wave32 waves
constexpr int MAX_PART   = 2512;  // >= ceil(300000/120) = 2500 ; LDS ~50KB/block

constexpr int SK_TPB    = 128;                      // score kernel: threads = rows/tile
constexpr int SK_ROWS   = 128;
constexpr int SK_DWORDS = SK_ROWS * NUM_CLASSES;    // 10368 dwords = 41.5KB LDS

// ---------------- CDNA5 async global->LDS staging ----------------
#if defined(__has_builtin)
# if __has_builtin(__builtin_amdgcn_global_load_async_to_lds_b128) && \
     __has_builtin(__builtin_amdgcn_global_load_async_to_lds_b32)
#  define HAVE_ASYNC_LDS 1
# endif
# if __has_builtin(__builtin_amdgcn_global_load_monitor_b32) && \
     __has_builtin(__builtin_amdgcn_s_monitor_sleep)
#  define HAVE_MONITOR 1
# endif
#endif

#ifndef __MEMORY_SCOPE_DEVICE
#define __MEMORY_SCOPE_DEVICE 1
#endif

typedef int v4i __attribute__((ext_vector_type(4)));

typedef __attribute__((address_space(1))) v4i* GlobV4;
typedef __attribute__((address_space(3))) v4i* LdsV4;
typedef __attribute__((address_space(1))) int* GlobI;
typedef __attribute__((address_space(3))) int* LdsI;

__device__ __forceinline__ void async_g2l_b128(const void* g, void* l) {
#ifdef HAVE_ASYNC_LDS
  __builtin_amdgcn_global_load_async_to_lds_b128((GlobV4)g, (LdsV4)l, 0, 0);
#else
  *(float4*)l = *(const float4*)g;
#endif
}

__device__ __forceinline__ void async_g2l_b32(const void* g, void* l) {
#ifdef HAVE_ASYNC_LDS
  __builtin_amdgcn_global_load_async_to_lds_b32((GlobI)g, (LdsI)l, 0, 0);
#else
  *(float*)l = *(const float*)g;
#endif
}

__device__ __forceinline__ void wait_async_zero() {
#ifdef HAVE_ASYNC_LDS
# if __has_builtin(__builtin_amdgcn_s_wait_asynccnt)
  __builtin_amdgcn_s_wait_asynccnt(0);
# else
  asm volatile("s_wait_asynccnt 0" ::: "memory");
# endif
#endif
}

// ---------------- agent-scope (device) memory helpers ----------------
__device__ __forceinline__ unsigned aload(const unsigned* p) {
  return __hip_atomic_load(p, __ATOMIC_RELAXED, __HIP_MEMORY_SCOPE_AGENT);
}
__device__ __forceinline__ void astore(unsigned* p, unsigned v) {
  __hip_atomic_store(p, v, __ATOMIC_RELAXED, __HIP_MEMORY_SCOPE_AGENT);
}

// ---------------- kernel 0: zero the grid-barrier counter ----------------
__global__ void init_counter(unsigned* counter) {
  __hip_atomic_store(counter, 0u, __ATOMIC_RELAXED, __HIP_MEMORY_SCOPE_AGENT);
}

// ---------------- kernel 1: per-box score = rowmax, NEG if label==0 ----------------
// Coalesced: async-copy a contiguous 128-row tile into LDS, reduce rows from LDS.
// label (first-occurrence argmax) > 0  <=>  row[0] < max(row)
__global__ __launch_bounds__(SK_TPB) void score_kernel(
    const float* __restrict__ cls, float* __restrict__ scores, int N) {
  __shared__ float s_tile[SK_DWORDS];

  const int tid = threadIdx.x;
  const long long row0    = (long long)blockIdx.x * SK_ROWS;
  const long long dw0     = row0 * NUM_CLASSES;
  const long long totalDw = (long long)N * NUM_CLASSES;

  // perfectly coalesced async global->LDS fill (consecutive lanes -> consecutive dwords)
  for (int i = tid; i < SK_DWORDS; i += SK_TPB) {
    long long g = dw0 + i;
    if (g >= totalDw) g = totalDw - 1;  // clamp; surplus data is ignored
    async_g2l_b32(cls + g, &s_tile[i]);
  }
  wait_async_zero();
  __syncthreads();

  const int row = (int)row0 + tid;
  if (row < N) {
    // row stride 81 dwords: 81*lane mod 64 banks is a permutation -> conflict-free
    const float* t = &s_tile[tid * NUM_CLASSES];
    float r0 = t[0];
    float m  = r0;
#pragma unroll
    for (int c = 1; c < NUM_CLASSES; ++c) m = fmaxf(m, t[c]);
    scores[row] = (r0 < m) ? m : NEGV;
  }
}

// ---------------- kernel 2: persistent LDS-resident greedy NMS ----------------
// parts layout: [2 parity slots][NMS_BLOCKS][8 dwords]:
//   {keyLo, keyHi, x1, y1, x2, y2, rawScoreBits, pad}
__global__ __launch_bounds__(NMS_TPB, 1) void nms_persistent(
    const float4* __restrict__ boxes, const float* __restrict__ cls,
    const float* __restrict__ scores, float* __restrict__ out,
    unsigned* __restrict__ counter, unsigned* __restrict__ parts, int N) {

  __shared__ float  s_sc[MAX_PART];
  __shared__ float4 s_box[MAX_PART];
  __shared__ unsigned long long s_red[NMS_TPB / 32];
  __shared__ unsigned long long s_winKey;
  __shared__ int   s_ownBlk;
  __shared__ float s_win[6];  // x1,y1,x2,y2, rawScore, gidx(bits)

  const int tid  = threadIdx.x;
  const int blk  = blockIdx.x;
  const int P    = (N + NMS_BLOCKS - 1) / NMS_BLOCKS;
  const int base = blk * P;
  int cnt = N - base;
  if (cnt > P) cnt = P;
  if (cnt < 0) cnt = 0;
  if (cnt > MAX_PART) cnt = MAX_PART;

  // ---- stage this block's slice into LDS via async global->LDS DMA ----
  for (int i = tid; i < cnt; i += NMS_TPB)
    async_g2l_b128(boxes + (base + i), &s_box[i]);
  const int n4 = cnt >> 2;
  for (int i = tid; i < n4; i += NMS_TPB)
    async_g2l_b128(scores + base + 4 * i, &s_sc[4 * i]);
  for (int i = 4 * n4 + tid; i < cnt; i += NMS_TPB)
    async_g2l_b32(scores + base + i, &s_sc[i]);
  wait_async_zero();
  __syncthreads();

  float wx1 = 0.f, wy1 = 0.f, wx2 = 0.f, wy2 = 0.f, warea = 0.f;
  bool have_prev = false;

  for (int it = 0; it < MAX_BOXES_K; ++it) {
    // ---- fused: suppress vs previous winner + local argmax (all in LDS) ----
    unsigned long long bestKey = 0ull;
    for (int i = tid; i < cnt; i += NMS_TPB) {
      float sc = s_sc[i];
      if (have_prev && sc > NEGV) {
        float4 b   = s_box[i];
        float xx1  = fmaxf(wx1, b.x), yy1 = fmaxf(wy1, b.y);
        float xx2  = fminf(wx2, b.z), yy2 = fminf(wy2, b.w);
        float inter = fmaxf(xx2 - xx1, 0.f) * fmaxf(yy2 - yy1, 0.f);
        float area  = fmaxf(b.z - b.x, 0.f) * fmaxf(b.w - b.y, 0.f);
        float uni   = fmaxf(warea + area - inter, 1e-8f);
        float iou   = inter / uni;
        if (iou > 0.4f) { sc = NEGV; s_sc[i] = sc; }
      }
      unsigned u  = __float_as_uint(sc);
      unsigned mk = (u & 0x80000000u) ? ~u : (u | 0x80000000u);  // monotone float key
      unsigned long long key =
          ((unsigned long long)mk << 32) | (0xFFFFFFFFu - (unsigned)(base + i));
      if (key > bestKey) bestKey = key;  // ties -> lowest index (reference argmax)
    }

    // ---- block-level argmax: wave32 shuffle + LDS ----
    unsigned long long k = bestKey;
#pragma unroll
    for (int off = 16; off > 0; off >>= 1) {
      unsigned long long o = __shfl_xor(k, off, 32);
      if (o > k) k = o;
    }
    if ((tid & 31) == 0) s_red[tid >> 5] = k;
    __syncthreads();

    unsigned* slot = parts + (unsigned)(it & 1) * (unsigned)(NMS_BLOCKS * 8);

    // ---- publish partial winner + grid barrier (monotone counter) ----
    if (tid == 0) {
      unsigned long long bk = 0ull;
#pragma unroll
      for (int w = 0; w < NMS_TPB / 32; ++w)
        if (s_red[w] > bk) bk = s_red[w];
      unsigned gi = 0xFFFFFFFFu - (unsigned)(bk & 0xFFFFFFFFu);
      int li = (int)gi - base;
      float4 bb = make_float4(0.f, 0.f, 0.f, 0.f);
      if (li >= 0 && li < cnt) bb = s_box[li];
      unsigned mk = (unsigned)(bk >> 32);
      unsigned su = (mk & 0x80000000u) ? (mk & 0x7FFFFFFFu) : ~mk;  // undo key map
      unsigned* p = slot + blk * 8;
      astore(p + 0, (unsigned)(bk & 0xFFFFFFFFu));
      astore(p + 1, mk);
      astore(p + 2, __float_as_uint(bb.x));
      astore(p + 3, __float_as_uint(bb.y));
      astore(p + 4, __float_as_uint(bb.z));
      astore(p + 5, __float_as_uint(bb.w));
      astore(p + 6, su);
      __hip_atomic_fetch_add(counter, 1u, __ATOMIC_RELEASE, __HIP_MEMORY_SCOPE_AGENT);
      unsigned target = (unsigned)(it + 1) * (unsigned)NMS_BLOCKS;
#ifdef HAVE_MONITOR
      // CDNA5 MWAIT: monitored device-scope load (reaches GL2 where the monitor tag
      // lives) + monitor-sleep; wakes when the counter cacheline is written.
      while ((unsigned)__builtin_amdgcn_global_load_monitor_b32(
                 (GlobI)counter, __ATOMIC_RELAXED, __MEMORY_SCOPE_DEVICE) < target)
        __builtin_amdgcn_s_monitor_sleep(1);
      (void)__hip_atomic_load(counter, __ATOMIC_ACQUIRE, __HIP_MEMORY_SCOPE_AGENT);
#else
      while (__hip_atomic_load(counter, __ATOMIC_ACQUIRE, __HIP_MEMORY_SCOPE_AGENT) < target)
        __builtin_amdgcn_s_sleep(2);
#endif
    }
    __syncthreads();

    // ---- every block redundantly reduces the 120 partials -> global winner ----
    unsigned long long pk = 0ull;
    if (tid < NMS_BLOCKS) {
      unsigned lo = aload(slot + tid * 8 + 0);
      unsigned hi = aload(slot + tid * 8 + 1);
      pk = ((unsigned long long)hi << 32) | lo;
    }
    unsigned long long kk = pk;
#pragma unroll
    for (int off = 16; off > 0; off >>= 1) {
      unsigned long long o = __shfl_xor(kk, off, 32);
      if (o > kk) kk = o;
    }
    if ((tid & 31) == 0) s_red[tid >> 5] = kk;
    __syncthreads();
    if (tid == 0) {
      unsigned long long wk = 0ull;
#pragma unroll
      for (int w = 0; w < NMS_TPB / 32; ++w)
        if (s_red[w] > wk) wk = s_red[w];
      s_winKey = wk;
      s_ownBlk = 0;
    }
    __syncthreads();
    if (tid < NMS_BLOCKS && pk == s_winKey) s_ownBlk = tid;  // keys unique (contain gidx)
    __syncthreads();
    if (tid == 0) {
      unsigned* wp = slot + s_ownBlk * 8;
      s_win[0] = __uint_as_float(aload(wp + 2));
      s_win[1] = __uint_as_float(aload(wp + 3));
      s_win[2] = __uint_as_float(aload(wp + 4));
      s_win[3] = __uint_as_float(aload(wp + 5));
      s_win[4] = __uint_as_float(aload(wp + 6));
      s_win[5] = __uint_as_float(0xFFFFFFFFu - (unsigned)(s_winKey & 0xFFFFFFFFu));
    }
    __syncthreads();

    wx1 = s_win[0]; wy1 = s_win[1]; wx2 = s_win[2]; wy2 = s_win[3];
    warea = fmaxf(wx2 - wx1, 0.f) * fmaxf(wy2 - wy1, 0.f);
    have_prev = true;

    // ---- block 0 emits this round's outputs ----
    if (blk == 0) {
      float wscore   = s_win[4];
      unsigned wgidx = __float_as_uint(s_win[5]);
      float scale    = (wscore > NEGV * 0.5f) ? 1.0f : 0.0f;  // valid = score > NEG/2
      if (tid < 4) out[it * 4 + tid] = s_win[tid] * scale;
      for (int c = tid; c < NUM_CLASSES; c += NMS_TPB)
        out[MAX_BOXES_K * 4 + it * NUM_CLASSES + c] =
            cls[(size_t)wgidx * NUM_CLASSES + c] * scale;
    }
    __syncthreads();  // protect s_red / s_win reuse next round
  }
}

// ---------------- launch ----------------
extern "C" void kernel_launch(void* const* d_in, const int* in_sizes, int n_in,
                              void* d_out, int out_size, void* d_ws, size_t ws_size,
                              hipStream_t stream) {
  (void)n_in; (void)out_size; (void)ws_size;
  const float* boxes = (const float*)d_in[0];  // (1, N, 4)
  const float* cls   = (const float*)d_in[1];  // (1, N, 81)
  const int N = in_sizes[0] / 4;

  char* ws            = (char*)d_ws;
  unsigned* counter   = (unsigned*)ws;          // [0, 64)
  unsigned* parts     = (unsigned*)(ws + 64);   // 2 * 120 * 8 * 4 = 7680 B
  float*    scores    = (float*)(ws + 8192);    // N * 4 B

  init_counter<<<1, 1, 0, stream>>>(counter);
  score_kernel<<<(N + SK_ROWS - 1) / SK_ROWS, SK_TPB, 0, stream>>>(cls, scores, N);
  nms_persistent<<<NMS_BLOCKS, NMS_TPB, 0, stream>>>(
      (const float4*)boxes, cls, scores, (float*)d_out, counter, parts, N);
}